// EmbeddingAttention_72507637891612
// MI455X (gfx1250) — compile-verified
//
#include <hip/hip_runtime.h>

// EmbeddingAttention fused kernel for gfx1250 (MI455X).
//   att_m[b,l] = mean_k <sent[b,idx[b,k],:], sent[b,l,:]>  = <eavg_b, row_l>
//   out[b,l,:] = sent[b,l,:] * att_m[b,l]
// Bandwidth-bound (~1 flop/byte); dot products batched 16 rows at a time with
// V_WMMA_F32_16X16X4_F32 (fp32-exact), accumulated over 32 K-slices of 4.

typedef float v2f __attribute__((ext_vector_type(2)));
typedef float v4f __attribute__((ext_vector_type(4)));
typedef float v8f __attribute__((ext_vector_type(8)));

constexpr int kB = 64;
constexpr int kL = 2048;
constexpr int kD = 128;
constexpr int kRowsPerBlock = 128;   // 8 waves x 16-row WMMA tiles
constexpr int kWaves = 8;

__global__ __launch_bounds__(256) void
embatt_kernel(const float* __restrict__ sent,
              const int* __restrict__ idxs,
              float* __restrict__ out)
{
    __shared__ float eavg[kD];            // 0.5*(e0+e1), folds the mean over k
    __shared__ float att[kRowsPerBlock];  // per-row attention scalars

    const int b       = blockIdx.y;
    const int rowBase = blockIdx.x * kRowsPerBlock;
    const int tid     = threadIdx.x;
    const int lane    = tid & 31;
    const int wave    = tid >> 5;

    const float* sb = sent + (size_t)b * kL * kD;

    // ---- Phase 0: averaged entity vector into LDS (one thread per d) ----
    if (tid < kD) {
        const int i0 = idxs[2 * b + 0];
        const int i1 = idxs[2 * b + 1];
        eavg[tid] = 0.5f * (sb[(size_t)i0 * kD + tid] + sb[(size_t)i1 * kD + tid]);
    }
    __syncthreads();

    // ---- Phase 1: 16 row-dots per wave via chained f32 WMMA ----
    // A (16x4 f32): lane t holds M = t%16, K = (t/16)*2 + {0,1}  -> float2 load
    // B (4x16 f32): lane t holds N = t%16, K = (t/16)*2 + {0,1}  -> eavg float2
    // D[m,n] = sum_k A[m,k]*B[k,n] = <row_m, eavg> for every n.
    const int m    = lane & 15;
    const int koff = (lane >> 4) << 1;
    const float* tile = sb + (size_t)(rowBase + wave * 16) * kD;

    v8f c = {0.f, 0.f, 0.f, 0.f, 0.f, 0.f, 0.f, 0.f};
#pragma unroll
    for (int d0 = 0; d0 < kD; d0 += 4) {
        v2f a  = *(const v2f*)(tile + (size_t)m * kD + d0 + koff);
        v2f bb = *(const v2f*)(&eavg[d0 + koff]);
        c = __builtin_amdgcn_wmma_f32_16x16x4_f32(
                /*neg_a=*/false, a, /*neg_b=*/false, bb,
                /*c_mod=*/(short)0, c, /*reuse_a=*/false, /*reuse_b=*/false);
    }

    // D layout: VGPR r on lane j holds M = r + 8*(j/16), N = j%16 (all N equal).
    // Lanes with N==0 (lanes 0 and 16) publish the 16 scalars to LDS.
    if ((lane & 15) == 0) {
        const int mhi = (lane >> 4) << 3;
#pragma unroll
        for (int r = 0; r < 8; ++r)
            att[wave * 16 + r + mhi] = c[r];
    }
    __syncthreads();

    // ---- Phase 2: coalesced scale+store (row = 32 lanes x float4) ----
    // Re-read hits WGP$/L2 (input fits in 192MB L2; just touched by phase 1).
    const size_t obase = (size_t)b * kL * kD + (size_t)rowBase * kD;
#pragma unroll 4
    for (int it = 0; it < kRowsPerBlock / kWaves; ++it) {
        const int rr  = it * kWaves + wave;
        const float s = att[rr];
        const float* src = sb + (size_t)(rowBase + rr) * kD + lane * 4;
        float*       dst = out + obase + (size_t)rr * kD + lane * 4;
        v4f v = *(const v4f*)src;
        v4f w = {v.x * s, v.y * s, v.z * s, v.w * s};
        *(v4f*)dst = w;
    }
}

extern "C" void kernel_launch(void* const* d_in, const int* in_sizes, int n_in,
                              void* d_out, int out_size, void* d_ws, size_t ws_size,
                              hipStream_t stream)
{
    const float* sent = (const float*)d_in[0];   // [B, L, D] fp32
    const int*   idxs = (const int*)d_in[1];     // [B, 2] int32
    float*       out  = (float*)d_out;           // [B, L, D] fp32

    dim3 grid(kL / kRowsPerBlock, kB);           // (16, 64) blocks
    embatt_kernel<<<grid, 256, 0, stream>>>(sent, idxs, out);
}